// Voxelization_89756226552188
// MI455X (gfx1250) — compile-verified
//
#include <hip/hip_runtime.h>
#include <stdint.h>

// Problem constants (match reference)
#define BATCH 8
#define NFEAT 64
#define NPTS  100000
#define RES   32
#define R3    (RES * RES * RES)            // 32768
#define GRID_ELEMS (BATCH * NFEAT * R3)    // 16,777,216 floats
#define EPSV  1e-6f
#define GROUPS (BATCH * NPTS / 4)          // 200,000 (N % 4 == 0, groups never cross batches)

#define AS1 __attribute__((address_space(1)))
#define AS3 __attribute__((address_space(3)))

typedef int v4i __attribute__((vector_size(16)));  // matches builtin's expected pointee type

#if defined(__has_builtin)
#if __has_builtin(__builtin_amdgcn_global_load_async_to_lds_b128) && \
    __has_builtin(__builtin_amdgcn_s_wait_asynccnt)
#define USE_ASYNC_LDS 1
#endif
#endif

__device__ __forceinline__ void atomic_add_f32(float* p, float v) {
  // relaxed agent-scope fadd -> global_atomic_add_f32 (no return, STOREcnt)
  __hip_atomic_fetch_add(p, v, __ATOMIC_RELAXED, __HIP_MEMORY_SCOPE_AGENT);
}

#ifdef USE_ASYNC_LDS
__device__ __forceinline__ void async_cp_b128(const float* gsrc, void* lds_dst) {
  __builtin_amdgcn_global_load_async_to_lds_b128(
      (AS1 v4i*)(uintptr_t)gsrc,                   // global src (AS1), 16B element
      (AS3 v4i*)(uint32_t)(uintptr_t)lds_dst,      // LDS dst (AS3): low 32 bits of generic LDS ptr
      0, 0);                                       // imm offset, imm cpol
}
#endif

// ---------------------------------------------------------------- zero grid+counts
__global__ __launch_bounds__(256) void k_zero(float* __restrict__ grid,
                                              float* __restrict__ counts) {
  int i = blockIdx.x * 256 + threadIdx.x;          // exactly GRID_ELEMS/4 threads
  float4 z = make_float4(0.f, 0.f, 0.f, 0.f);
  ((float4*)grid)[i] = z;
  if (i < (BATCH * R3) / 4) ((float4*)counts)[i] = z;
}

// ---------------------------------------------------------------- per-batch mean + radius
__global__ __launch_bounds__(256) void k_stats(const float* __restrict__ coords,
                                               float4* __restrict__ stats) {
  const int b = blockIdx.x, t = threadIdx.x;
  const float* cb = coords + (size_t)b * 3 * NPTS;
  __shared__ float sx[256], sy[256], sz[256];
  float ax = 0.f, ay = 0.f, az = 0.f;
  for (int n = t; n < NPTS; n += 256) {
    ax += cb[n]; ay += cb[NPTS + n]; az += cb[2 * NPTS + n];
  }
  sx[t] = ax; sy[t] = ay; sz[t] = az;
  __syncthreads();
  for (int s = 128; s > 0; s >>= 1) {
    if (t < s) { sx[t] += sx[t + s]; sy[t] += sy[t + s]; sz[t] += sz[t + s]; }
    __syncthreads();
  }
  __shared__ float mx, my, mz;
  if (t == 0) {
    const float invn = 1.f / (float)NPTS;
    mx = sx[0] * invn; my = sy[0] * invn; mz = sz[0] * invn;
  }
  __syncthreads();
  float m = 0.f;
  for (int n = t; n < NPTS; n += 256) {
    float dx = cb[n] - mx, dy = cb[NPTS + n] - my, dz = cb[2 * NPTS + n] - mz;
    m = fmaxf(m, dx * dx + dy * dy + dz * dz);
  }
  sx[t] = m;
  __syncthreads();
  for (int s = 128; s > 0; s >>= 1) {
    if (t < s) sx[t] = fmaxf(sx[t], sx[t + s]);
    __syncthreads();
  }
  if (t == 0) stats[b] = make_float4(mx, my, mz, sqrtf(sx[0]));
}

// ---------------------------------------------------------------- norm_coords + counts
__global__ __launch_bounds__(256) void k_points(const float* __restrict__ coords,
                                                const float4* __restrict__ stats,
                                                float* __restrict__ norm_out,
                                                float* __restrict__ counts) {
  int i = blockIdx.x * 256 + threadIdx.x;          // exactly BATCH*NPTS threads
  int b = i / NPTS, n = i - b * NPTS;
  float4 st = stats[b];
  float inv = 1.f / (2.f * st.w + EPSV);
  const float* cb = coords + (size_t)b * 3 * NPTS + n;
  float nx = ((cb[0]            - st.x) * inv + 0.5f) * (float)RES;
  float ny = ((cb[NPTS]         - st.y) * inv + 0.5f) * (float)RES;
  float nz = ((cb[2 * NPTS]     - st.z) * inv + 0.5f) * (float)RES;
  nx = fminf(fmaxf(nx, 0.f), (float)(RES - 1));
  ny = fminf(fmaxf(ny, 0.f), (float)(RES - 1));
  nz = fminf(fmaxf(nz, 0.f), (float)(RES - 1));
  float* nb = norm_out + (size_t)b * 3 * NPTS + n;
  nb[0] = nx; nb[NPTS] = ny; nb[2 * NPTS] = nz;
  int vox = ((int)rintf(nx) * RES + (int)rintf(ny)) * RES + (int)rintf(nz);  // RNE like jnp.round
  atomic_add_f32(&counts[b * R3 + vox], 1.f);
}

// ---------------------------------------------------------------- feature scatter-add
__global__ __launch_bounds__(256) void k_scatter(const float* __restrict__ features,
                                                 const float* __restrict__ norm_in,
                                                 float* __restrict__ grid) {
#ifdef USE_ASYNC_LDS
  __shared__ float4 stage[2][256];                 // 8 KB double buffer, per-lane slots
#endif
  int g = blockIdx.x * 256 + threadIdx.x;
  if (g >= GROUPS) return;
  int p = g * 4;
  int b = p / NPTS, n = p - b * NPTS;
  const float* nb = norm_in + (size_t)b * 3 * NPTS + n;
  float4 nx = *(const float4*)(nb);
  float4 ny = *(const float4*)(nb + NPTS);
  float4 nz = *(const float4*)(nb + 2 * NPTS);
  int idx0 = ((int)rintf(nx.x) * RES + (int)rintf(ny.x)) * RES + (int)rintf(nz.x);
  int idx1 = ((int)rintf(nx.y) * RES + (int)rintf(ny.y)) * RES + (int)rintf(nz.y);
  int idx2 = ((int)rintf(nx.z) * RES + (int)rintf(ny.z)) * RES + (int)rintf(nz.z);
  int idx3 = ((int)rintf(nx.w) * RES + (int)rintf(ny.w)) * RES + (int)rintf(nz.w);
  const float* fb = features + (size_t)b * NFEAT * NPTS + n;
  float* gb = grid + (size_t)b * NFEAT * R3;

#ifdef USE_ASYNC_LDS
  const int t = threadIdx.x;
  async_cp_b128(fb, &stage[0][t]);                 // stage feature row 0
  for (int f = 0; f < NFEAT - 1; ++f) {
    async_cp_b128(fb + (size_t)(f + 1) * NPTS, &stage[(f + 1) & 1][t]);
    __builtin_amdgcn_s_wait_asynccnt(1);           // row f arrived (in-order completion)
    float4 v = stage[f & 1][t];
    float* gp = gb + f * R3;
    atomic_add_f32(gp + idx0, v.x);
    atomic_add_f32(gp + idx1, v.y);
    atomic_add_f32(gp + idx2, v.z);
    atomic_add_f32(gp + idx3, v.w);
  }
  __builtin_amdgcn_s_wait_asynccnt(0);
  {
    float4 v = stage[(NFEAT - 1) & 1][t];
    float* gp = gb + (NFEAT - 1) * R3;
    atomic_add_f32(gp + idx0, v.x);
    atomic_add_f32(gp + idx1, v.y);
    atomic_add_f32(gp + idx2, v.z);
    atomic_add_f32(gp + idx3, v.w);
  }
#else
  for (int f = 0; f < NFEAT; ++f) {
    const float* fp = fb + (size_t)f * NPTS;
    if (f + 1 < NFEAT) __builtin_prefetch(fp + NPTS, 0, 0);  // global_prefetch_b8
    float4 v = *(const float4*)fp;
    float* gp = gb + f * R3;
    atomic_add_f32(gp + idx0, v.x);
    atomic_add_f32(gp + idx1, v.y);
    atomic_add_f32(gp + idx2, v.z);
    atomic_add_f32(gp + idx3, v.w);
  }
#endif
}

// ---------------------------------------------------------------- divide by counts
// Empty-voxel fast path: k_zero guarantees untouched voxels are exactly 0 and they
// receive no atomics, so a float4 of voxels with all counts==0 needs no load/store
// across all 64 feature planes. Gaussian clouds leave most of the 32^3 grid empty
// and occupancy is spatially clustered, so whole waves take the skip together.
__global__ __launch_bounds__(256) void k_divide(const float* __restrict__ counts,
                                                float* __restrict__ grid) {
  int t = blockIdx.x * 256 + threadIdx.x;          // exactly BATCH*R3/4 threads
  int b = t / (R3 / 4);
  int v = (t - b * (R3 / 4)) * 4;
  float4 c = *(const float4*)(counts + b * R3 + v);
  if (c.x == 0.f && c.y == 0.f && c.z == 0.f && c.w == 0.f) return;
  float4 inv;
  inv.x = 1.f / fmaxf(c.x, 1.f);
  inv.y = 1.f / fmaxf(c.y, 1.f);
  inv.z = 1.f / fmaxf(c.z, 1.f);
  inv.w = 1.f / fmaxf(c.w, 1.f);
  float* gb = grid + (size_t)b * NFEAT * R3 + v;
  for (int f = 0; f < NFEAT; ++f) {
    float4* p = (float4*)(gb + (size_t)f * R3);
    float4 gg = *p;
    gg.x *= inv.x; gg.y *= inv.y; gg.z *= inv.z; gg.w *= inv.w;
    *p = gg;
  }
}

extern "C" void kernel_launch(void* const* d_in, const int* in_sizes, int n_in,
                              void* d_out, int out_size, void* d_ws, size_t ws_size,
                              hipStream_t stream) {
  const float* features = (const float*)d_in[0];   // [B,F,N]
  const float* coords   = (const float*)d_in[1];   // [B,3,N]
  float* grid     = (float*)d_out;                 // [B,F,32,32,32]
  float* norm_out = grid + GRID_ELEMS;             // [B,3,N]
  // workspace: [0,256B) per-batch stats float4; [256B, 256B+1MB) counts
  float4* stats = (float4*)d_ws;
  float*  counts = (float*)d_ws + 64;

  k_zero<<<GRID_ELEMS / 4 / 256, 256, 0, stream>>>(grid, counts);
  k_stats<<<BATCH, 256, 0, stream>>>(coords, stats);
  k_points<<<(BATCH * NPTS) / 256, 256, 0, stream>>>(coords, stats, norm_out, counts);
  k_scatter<<<(GROUPS + 255) / 256, 256, 0, stream>>>(features, norm_out, grid);
  k_divide<<<(BATCH * R3 / 4) / 256, 256, 0, stream>>>(counts, grid);
}